// GCN_65558380806313
// MI455X (gfx1250) — compile-verified
//
#include <hip/hip_runtime.h>

typedef float v2f __attribute__((ext_vector_type(2)));
typedef float v8f __attribute__((ext_vector_type(8)));

// D = A(16x4 f32) x B(4x16 f32) + C(16x16 f32)
#define WMMA_F32X4(a, b, c) \
  __builtin_amdgcn_wmma_f32_16x16x4_f32(false, (a), false, (b), (short)0, (c), false, false)

// ---------------------------------------------------------------------------
// degree / normalization kernels
// ---------------------------------------------------------------------------
__global__ void k_deg_init(float* __restrict__ deg, int n) {
  int i = blockIdx.x * blockDim.x + threadIdx.x;
  if (i < n) deg[i] = 1.0f;  // self loop contributes 1
}

__global__ void k_deg_edges(const long long* __restrict__ row, float* __restrict__ deg, int E) {
  int e = blockIdx.x * blockDim.x + threadIdx.x;
  if (e < E) atomicAdd(&deg[(int)row[e]], 1.0f);
}

__global__ void k_deg_rsqrt(float* __restrict__ d, int n) {
  int i = blockIdx.x * blockDim.x + threadIdx.x;
  if (i < n) d[i] = rsqrtf(d[i]);  // deg >= 1 always
}

// ---------------------------------------------------------------------------
// fp32 WMMA GEMM: H[N x 64] = A[N x 64] @ W[64 x 64]
// one wave per 16x16 output tile; K = 64 via 16 x wmma_f32_16x16x4_f32
// ---------------------------------------------------------------------------
__global__ void __launch_bounds__(32) k_gemm_k64(const float* __restrict__ A,
                                                 const float* __restrict__ W,
                                                 float* __restrict__ H, int nRows) {
  const int lane = threadIdx.x & 31;
  const int half = lane >> 4;    // 0/1 selects K pair
  const int lm   = lane & 15;    // M (for A) / N (for B,C)
  const int R = blockIdx.x * 16;
  const int C = blockIdx.y * 16;
  const int ra = min(R + lm, nRows - 1);  // clamp A-row reads for safety

  v8f acc = {};
#pragma unroll
  for (int k = 0; k < 64; k += 4) {
    // A frag: lane lm holds row M=lm, K = k + 2*half .. +1  (contiguous -> v2f)
    v2f a = *(const v2f*)(A + (size_t)ra * 64 + k + 2 * half);
    // B frag: VGPR0 = row K=k+2*half, VGPR1 = row K=k+2*half+1, N = C+lm
    v2f b;
    b.x = W[(k + 2 * half) * 64 + C + lm];
    b.y = W[(k + 2 * half + 1) * 64 + C + lm];
    acc = WMMA_F32X4(a, b, acc);
  }
#pragma unroll
  for (int r = 0; r < 8; ++r) {
    int rowIdx = R + r + 8 * half;
    if (rowIdx < nRows) H[(size_t)rowIdx * 64 + C + lm] = acc[r];
  }
}

// ---------------------------------------------------------------------------
// agg init: self-loop (weight dis[i]^2) + bias
// ---------------------------------------------------------------------------
__global__ void k_agg_init(const float* __restrict__ h, const float* __restrict__ dis,
                           const float* __restrict__ bias, float* __restrict__ agg, int n64) {
  int t = blockIdx.x * blockDim.x + threadIdx.x;
  if (t < n64) {
    int i = t >> 6, f = t & 63;
    float di = dis[i];
    agg[t] = di * di * h[t] + bias[f];
  }
}

// ---------------------------------------------------------------------------
// edge scatter: one wave32 per edge, each lane handles 2 features
// agg[row] += (dis[row]*ew*dis[col]) * h[col]
// ---------------------------------------------------------------------------
__global__ void k_agg_edges(const long long* __restrict__ row, const long long* __restrict__ col,
                            const float* __restrict__ ew, const float* __restrict__ dis,
                            const float* __restrict__ h, float* __restrict__ agg, int E) {
  long long t = (long long)blockIdx.x * blockDim.x + threadIdx.x;
  int e = (int)(t >> 5);
  int lane = (int)(t & 31);
  if (e >= E) return;
  int r = (int)row[e];
  int c = (int)col[e];
  float w = dis[r] * ew[e] * dis[c];
  v2f hv = *(const v2f*)(h + (size_t)c * 64 + lane * 2);
  float* dst = agg + (size_t)r * 64 + lane * 2;
  atomicAdd(dst, w * hv.x);
  atomicAdd(dst + 1, w * hv.y);
}

__global__ void k_relu(float* __restrict__ p, int n) {
  int i = blockIdx.x * blockDim.x + threadIdx.x;
  if (i < n) p[i] = fmaxf(p[i], 0.0f);
}

// ---------------------------------------------------------------------------
// fused FC head: out[16x10] = relu(A[16x64] @ w1[64x128] + b1) @ w2[128x10] + b2
// one wave per 16-node tile; fc1 tile buffered in LDS (16x128 = 8KB)
// ---------------------------------------------------------------------------
__global__ void __launch_bounds__(32) k_fc_head(const float* __restrict__ A,
                                                const float* __restrict__ w1,
                                                const float* __restrict__ b1,
                                                const float* __restrict__ w2,
                                                const float* __restrict__ b2,
                                                float* __restrict__ out, int nRows) {
  __shared__ float F[16 * 128];
  const int lane = threadIdx.x & 31;
  const int half = lane >> 4;
  const int lm   = lane & 15;
  const int R = blockIdx.x * 16;
  const int ra = min(R + lm, nRows - 1);

  // ---- fc1 + relu -> LDS ----
#pragma unroll
  for (int ct = 0; ct < 8; ++ct) {
    const int C = ct * 16;
    v8f acc = {};
#pragma unroll
    for (int k = 0; k < 64; k += 4) {
      v2f a = *(const v2f*)(A + (size_t)ra * 64 + k + 2 * half);
      v2f b;
      b.x = w1[(k + 2 * half) * 128 + C + lm];
      b.y = w1[(k + 2 * half + 1) * 128 + C + lm];
      acc = WMMA_F32X4(a, b, acc);
    }
    float bias = b1[C + lm];
#pragma unroll
    for (int r = 0; r < 8; ++r)
      F[(r + 8 * half) * 128 + C + lm] = fmaxf(acc[r] + bias, 0.0f);
  }
  __syncthreads();

  // ---- fc2: 16x10 (padded to 16x16 tile, cols >= 10 masked) ----
  v8f acc = {};
#pragma unroll
  for (int k = 0; k < 128; k += 4) {
    v2f a;
    a.x = F[lm * 128 + k + 2 * half];
    a.y = F[lm * 128 + k + 2 * half + 1];
    v2f b;
    b.x = (lm < 10) ? w2[(k + 2 * half) * 10 + lm] : 0.0f;
    b.y = (lm < 10) ? w2[(k + 2 * half + 1) * 10 + lm] : 0.0f;
    acc = WMMA_F32X4(a, b, acc);
  }
  if (lm < 10) {
    float bias = b2[lm];
#pragma unroll
    for (int r = 0; r < 8; ++r) {
      int rowIdx = R + r + 8 * half;
      if (rowIdx < nRows) out[(size_t)rowIdx * 10 + lm] = acc[r] + bias;
    }
  }
}

// ---------------------------------------------------------------------------
extern "C" void kernel_launch(void* const* d_in, const int* in_sizes, int n_in,
                              void* d_out, int out_size, void* d_ws, size_t ws_size,
                              hipStream_t stream) {
  const float*     x    = (const float*)d_in[0];
  const long long* ei   = (const long long*)d_in[1];
  const float*     ea   = (const float*)d_in[2];
  const float*     W1   = (const float*)d_in[3];
  const float*     b1   = (const float*)d_in[4];
  const float*     W2   = (const float*)d_in[5];
  const float*     b2   = (const float*)d_in[6];
  const float*     fc1w = (const float*)d_in[7];
  const float*     fc1b = (const float*)d_in[8];
  const float*     fc2w = (const float*)d_in[9];
  const float*     fc2b = (const float*)d_in[10];
  float* out = (float*)d_out;

  const int N = in_sizes[0] / 64;
  const int E = in_sizes[1] / 2;
  const long long* row = ei;
  const long long* col = ei + E;

  float* ws   = (float*)d_ws;
  float* dis  = ws;                        // N
  float* bufA = ws + N;                    // N*64  (GEMM outputs H1/H2)
  float* bufB = bufA + (size_t)N * 64;     // N*64  (aggregated A1/A2)

  const int TB = 256;
  const int n64 = N * 64;
  const int rowTiles = (N + 15) / 16;
  const long long edgeThreads = (long long)E * 32;
  const int edgeBlocks = (int)((edgeThreads + TB - 1) / TB);

  // degree normalization: deg = 1 + sum_edges, dis = deg^-1/2
  k_deg_init<<<(N + TB - 1) / TB, TB, 0, stream>>>(dis, N);
  k_deg_edges<<<(E + TB - 1) / TB, TB, 0, stream>>>(row, dis, E);
  k_deg_rsqrt<<<(N + TB - 1) / TB, TB, 0, stream>>>(dis, N);

  // ---- GCN layer 1: bufB = relu(agg(x @ W1) + b1) ----
  k_gemm_k64<<<dim3(rowTiles, 4), 32, 0, stream>>>(x, W1, bufA, N);
  k_agg_init<<<(n64 + TB - 1) / TB, TB, 0, stream>>>(bufA, dis, b1, bufB, n64);
  k_agg_edges<<<edgeBlocks, TB, 0, stream>>>(row, col, ea, dis, bufA, bufB, E);
  k_relu<<<(n64 + TB - 1) / TB, TB, 0, stream>>>(bufB, n64);

  // ---- GCN layer 2: bufB = agg(bufB @ W2) + b2  (no relu) ----
  k_gemm_k64<<<dim3(rowTiles, 4), 32, 0, stream>>>(bufB, W2, bufA, N);
  k_agg_init<<<(n64 + TB - 1) / TB, TB, 0, stream>>>(bufA, dis, b2, bufB, n64);
  k_agg_edges<<<edgeBlocks, TB, 0, stream>>>(row, col, ea, dis, bufA, bufB, E);

  // ---- FC head (fused fc1 + relu + fc2) ----
  k_fc_head<<<rowTiles, 32, 0, stream>>>(bufB, fc1w, fc1b, fc2w, fc2b, out, N);
}